// WavetableSynthV2_72224170050168
// MI455X (gfx1250) — compile-verified
//
#include <hip/hip_runtime.h>

// Problem constants (from reference)
#define SR_F      16000.0f
#define BS        32
#define N_WT      20
#define WT_LEN    512
#define T_LEN     64000

// Scan tiling
#define BLOCK     256
#define ITEMS     16
#define CHUNK     (BLOCK * ITEMS)                    // 4096
#define NCHUNK    ((T_LEN + CHUNK - 1) / CHUNK)      // 16

typedef float v2f __attribute__((ext_vector_type(2)));
typedef float v8f __attribute__((ext_vector_type(8)));

// ---------------------------------------------------------------------------
// Phase 0: mixed[b, k] = sum_n attention[b,n] * wavetables[b,n,k]
// One wave handles one (batch, 16-column tile) via V_WMMA_F32_16X16X4_F32.
// A (16x4 f32): M=0 row = attention slice, rows 1..15 zero.
//   lane 0  holds A[0][k0+0], A[0][k0+1]; lane 16 holds A[0][k0+2], A[0][k0+3].
// B (4x16 f32): VGPR0 lanes0-15 = row k0+0, lanes16-31 = row k0+2;
//               VGPR1 lanes0-15 = row k0+1, lanes16-31 = row k0+3.
// D (16x16 f32): VGPR0 lanes0-15 = row M=0 -> the mixed values we want.
// Also copies attention to the output tail (tuple return: (out, attention)).
// ---------------------------------------------------------------------------
__global__ void mix_wmma_kernel(const float* __restrict__ wavetables,
                                const float* __restrict__ attention,
                                float* __restrict__ mixed,
                                float* __restrict__ att_out) {
    const int lane   = threadIdx.x & 31;
    const int wib    = threadIdx.x >> 5;
    const int gwave  = blockIdx.x * (blockDim.x >> 5) + wib;  // [0, BS*32)
    const int b      = gwave >> 5;         // batch
    const int j      = gwave & 31;         // 16-col tile within WT_LEN

    const float* wt  = wavetables + (size_t)b * N_WT * WT_LEN;
    const float* att = attention  + b * N_WT;

    const int col  = j * 16 + (lane & 15);
    const int krow = (lane < 16) ? 0 : 2;

    v8f c = {};  // fp32 accumulator, starts at 0
#pragma unroll
    for (int s = 0; s < 5; ++s) {          // K = 20 = 5 * 4
        const int k0 = 4 * s;
        v2f a; a.x = 0.0f; a.y = 0.0f;
        if (lane == 0)  { a.x = att[k0 + 0]; a.y = att[k0 + 1]; }
        if (lane == 16) { a.x = att[k0 + 2]; a.y = att[k0 + 3]; }
        v2f bm;
        bm.x = wt[(k0 + krow + 0) * WT_LEN + col];
        bm.y = wt[(k0 + krow + 1) * WT_LEN + col];
        c = __builtin_amdgcn_wmma_f32_16x16x4_f32(
                /*neg_a=*/false, a, /*neg_b=*/false, bm,
                /*c_mod=*/(short)0, c, /*reuse_a=*/false, /*reuse_b=*/false);
    }
    if (lane < 16) {
        mixed[(size_t)b * WT_LEN + j * 16 + lane] = c[0];  // D row M=0
    }

    // second output of the reference tuple: attention, copied verbatim
    const int tid = blockIdx.x * blockDim.x + threadIdx.x;
    if (tid < BS * N_WT) att_out[tid] = attention[tid];
}

// ---------------------------------------------------------------------------
// Phase 1: per-(row, chunk) sum of increments -> partials[b][chunk]
// ---------------------------------------------------------------------------
__global__ void chunk_sums_kernel(const float* __restrict__ pitch,
                                  float* __restrict__ partials) {
    const int chunk = blockIdx.x;
    const int b     = blockIdx.y;
    const float* p  = pitch + (size_t)b * T_LEN;
    const int t0    = chunk * CHUNK + threadIdx.x * ITEMS;

    float s = 0.0f;
#pragma unroll
    for (int i = 0; i < ITEMS; ++i) {
        const int t = t0 + i;
        if (t < T_LEN) s += p[t] * (1.0f / SR_F) * (float)WT_LEN;
    }

    __shared__ float red[BLOCK];
    red[threadIdx.x] = s;
    __syncthreads();
    for (int off = BLOCK / 2; off > 0; off >>= 1) {
        if (threadIdx.x < off) red[threadIdx.x] += red[threadIdx.x + off];
        __syncthreads();
    }
    if (threadIdx.x == 0) partials[b * NCHUNK + chunk] = red[0];
}

// ---------------------------------------------------------------------------
// Phase 2: tiny in-place exclusive scan of the NCHUNK partials per row
// ---------------------------------------------------------------------------
__global__ void chunk_offsets_kernel(float* partials) {
    const int b = threadIdx.x;
    if (b < BS) {
        float run = 0.0f;
        for (int c = 0; c < NCHUNK; ++c) {
            const float v = partials[b * NCHUNK + c];
            partials[b * NCHUNK + c] = run;
            run += v;
        }
    }
}

// ---------------------------------------------------------------------------
// Phase 3: in-chunk scan + wrap + LDS-table lerp + amplitude scale
// ---------------------------------------------------------------------------
__global__ void synth_kernel(const float* __restrict__ pitch,
                             const float* __restrict__ amplitude,
                             const float* __restrict__ mixed,
                             const float* __restrict__ partials,
                             float* __restrict__ out) {
    const int chunk = blockIdx.x;
    const int b     = blockIdx.y;

    __shared__ float table[WT_LEN];
    __shared__ float scan[BLOCK];

    for (int i = threadIdx.x; i < WT_LEN; i += BLOCK)
        table[i] = mixed[(size_t)b * WT_LEN + i];

    const float* p  = pitch + (size_t)b * T_LEN;
    const float* p1 = pitch + (size_t)1 * T_LEN;   // "row 1" bug, replicated
    const int t0    = chunk * CHUNK + threadIdx.x * ITEMS;

    float inc[ITEMS];
    float run = 0.0f;
#pragma unroll
    for (int i = 0; i < ITEMS; ++i) {
        const int t = t0 + i;
        const float v = (t < T_LEN) ? p[t] * (1.0f / SR_F) * (float)WT_LEN : 0.0f;
        run += v;
        inc[i] = run;          // inclusive sum within this thread's items
    }

    scan[threadIdx.x] = run;
    __syncthreads();
    // Hillis-Steele inclusive scan over BLOCK thread totals
    for (int off = 1; off < BLOCK; off <<= 1) {
        const float mine = scan[threadIdx.x];
        const float add  = (threadIdx.x >= off) ? scan[threadIdx.x - off] : 0.0f;
        __syncthreads();
        scan[threadIdx.x] = mine + add;
        __syncthreads();
    }
    const float excl = (threadIdx.x == 0) ? 0.0f : scan[threadIdx.x - 1];
    const float base = partials[b * NCHUNK + chunk] + excl;

#pragma unroll
    for (int i = 0; i < ITEMS; ++i) {
        const int t = t0 + i;
        if (t >= T_LEN) break;
        const float S    = base + inc[i];
        const float inc1 = p1[t] * (1.0f / SR_F) * (float)WT_LEN;
        float m = fmodf(S - inc1, (float)WT_LEN);
        if (m < 0.0f)              m += (float)WT_LEN;   // python-style mod
        if (m >= (float)WT_LEN)    m  = 0.0f;            // rounding guard
        const float fl    = floorf(m);
        const float alpha = m - fl;
        int il = (int)fl;          if (il > WT_LEN - 1) il = WT_LEN - 1;
        int ih = (int)ceilf(m);    ih &= (WT_LEN - 1);   // ceil can hit WT_LEN
        const float wl  = table[il];
        const float wh  = table[ih];
        const float sig = wl + alpha * (wh - wl);
        out[(size_t)b * T_LEN + t] = sig * amplitude[(size_t)b * T_LEN + t];
    }
}

// ---------------------------------------------------------------------------
extern "C" void kernel_launch(void* const* d_in, const int* in_sizes, int n_in,
                              void* d_out, int out_size, void* d_ws, size_t ws_size,
                              hipStream_t stream) {
    const float* pitch      = (const float*)d_in[0];   // (BS, T)
    const float* amplitude  = (const float*)d_in[1];   // (BS, T, 1)
    const float* wavetables = (const float*)d_in[2];   // (BS, N_WT, WT_LEN)
    const float* attention  = (const float*)d_in[3];   // (BS, N_WT)

    float* out      = (float*)d_out;                   // (BS, T) then (BS, N_WT)
    float* att_out  = out + (size_t)BS * T_LEN;

    float* mixed    = (float*)d_ws;                    // BS * WT_LEN floats
    float* partials = mixed + BS * WT_LEN;             // BS * NCHUNK floats

    // Phase 0: WMMA attention-mix of wavetables (+ attention passthrough)
    {
        const int waves  = BS * (WT_LEN / 16);         // 1024 waves
        const int blocks = waves / (BLOCK / 32);       // 128 blocks
        mix_wmma_kernel<<<blocks, BLOCK, 0, stream>>>(wavetables, attention,
                                                      mixed, att_out);
    }
    // Phase 1: chunk sums
    chunk_sums_kernel<<<dim3(NCHUNK, BS), BLOCK, 0, stream>>>(pitch, partials);
    // Phase 2: per-row exclusive scan of chunk sums
    chunk_offsets_kernel<<<1, 32, 0, stream>>>(partials);
    // Phase 3: scan + gather + scale
    synth_kernel<<<dim3(NCHUNK, BS), BLOCK, 0, stream>>>(pitch, amplitude,
                                                         mixed, partials, out);
}